// CoTGlow_18760417149534
// MI455X (gfx1250) — compile-verified
//
#include <hip/hip_runtime.h>

// ---------------------------------------------------------------------------
// CoTGlow forward_vector_field_half, algebraically reduced to per-scale small
// grids (periodic-tiling of the FFT of zero-upsampled latents + alias-folding
// of alpha_c onto the coarse grid). f32 WMMA (V_WMMA_F32_16X16X4_F32) does the
// channel-mixing GEMM AND the column inverse-DFT (as a real GEMM with
// interleaved cos/-sin twiddle rows). Row IDFT + small scales stay on VALU.
// ---------------------------------------------------------------------------

typedef float v2f __attribute__((ext_vector_type(2)));
typedef float v8f __attribute__((ext_vector_type(8)));

#define TWO_PI_OVER_32 0.19634954084936207f

// per-scale tables: channels, h(=w), A column offset, l-chunk width
__device__ constexpr int SC_C[6]   = {384, 96, 48, 24, 12, 6};
__device__ constexpr int SC_H[6]   = {1, 2, 4, 8, 16, 32};
__device__ constexpr int SC_OFF[6] = {0, 384, 480, 528, 552, 564};
__device__ constexpr int SC_LC[6]  = {1, 2, 4, 8, 8, 4};

// ahat: [scale][b][ch][k*w+l][2] floats, complex elem counts 32*c*hw
__device__ constexpr int AH_CUM[6]    = {0, 12288, 24576, 49152, 98304, 196608};
__device__ constexpr int AHAT_BASE[6] = {0, 24576, 49152, 98304, 196608, 393216};
#define AHAT_TOTAL_CPLX 393216
#define AHAT_TOTAL_FLOATS 786432

// beta: [scale][m][k*w+l][2] floats, complex elem counts 1024*hw
__device__ constexpr int BT_CUM[6]    = {0, 1024, 5120, 21504, 87040, 349184};
__device__ constexpr int BETA_BASE[6] = {0, 2048, 10240, 43008, 174080, 698368};
#define BETA_TOTAL_CPLX 1397760

// ---------------------------------------------------------------------------
// Kernel 1: ahat[b,ch,k,l] = sum_{y,x} a[b,ch,y,x] * e^{-2pi i (k y/h + l x/w)}
// ---------------------------------------------------------------------------
__global__ void dft2_kernel(const float* __restrict__ a0, const float* __restrict__ a1,
                            const float* __restrict__ a2, const float* __restrict__ a3,
                            const float* __restrict__ a4, const float* __restrict__ a5,
                            float* __restrict__ ahat) {
    __shared__ float sC[32], sS[32];
    const int t = threadIdx.x;
    if (t < 32) {
        float ang = (float)t * TWO_PI_OVER_32;
        sC[t] = cosf(ang);
        sS[t] = sinf(ang);
    }
    __syncthreads();
    int gid = blockIdx.x * 256 + t;
    if (gid >= AHAT_TOTAL_CPLX) return;

    const float* aps[6] = {a0, a1, a2, a3, a4, a5};
    int i = 0;
#pragma unroll
    for (int s = 1; s < 6; s++)
        if (gid >= AH_CUM[s]) i = s;
    int idx = gid - AH_CUM[i];

    const int c = SC_C[i], h = SC_H[i], w = h, hw = h * w, rh = 32 / h;
    int b = idx / (c * hw);
    int rem = idx - b * (c * hw);
    int ch = rem / hw;
    int p = rem - ch * hw;
    int k = p / w, l = p - (p / w) * w;

    const float* ap = aps[i] + (size_t)(b * c + ch) * hw;
    float re = 0.f, im = 0.f;
    for (int y = 0; y < h; y++) {
        int ky = k * y * rh;
        for (int x = 0; x < w; x++) {
            float v = ap[y * w + x];
            int tw = (ky + l * x * rh) & 31;
            re += v * sC[tw];
            im -= v * sS[tw];
        }
    }
    float* o = ahat + AHAT_BASE[i] + (size_t)idx * 2;
    o[0] = re;
    o[1] = im;
}

// ---------------------------------------------------------------------------
// Kernel 2: beta_i[m,k,l] = sum_{j<rh,j2<rw} alpha_c[m, k+j*h, l+j2*w]
// alpha_c[m,K,L] = alpha[m,K,L] - i*alpha[m,(32-K)%32,(32-L)%32]
// ---------------------------------------------------------------------------
__global__ void beta_kernel(const float* __restrict__ alpha, float* __restrict__ beta) {
    int gid = blockIdx.x * 256 + threadIdx.x;
    if (gid >= BETA_TOTAL_CPLX) return;
    int i = 0;
#pragma unroll
    for (int s = 1; s < 6; s++)
        if (gid >= BT_CUM[s]) i = s;
    int idx = gid - BT_CUM[i];

    const int h = SC_H[i], w = h, hw = h * w, rh = 32 / h;
    int m = idx / hw;
    int p = idx - m * hw;
    int k = p / w, l = p - (p / w) * w;

    const float* am = alpha + (size_t)m * 1024;
    float re = 0.f, im = 0.f;
    for (int j = 0; j < rh; j++) {
        int K = k + j * h;
        int Kr = (32 - K) & 31;
        for (int j2 = 0; j2 < rh; j2++) {
            int L = l + j2 * w;
            int Lr = (32 - L) & 31;
            re += am[K * 32 + L];
            im -= am[Kr * 32 + Lr];
        }
    }
    float* o = beta + BETA_BASE[i] + (size_t)idx * 2;
    o[0] = re;
    o[1] = im;
}

// ---------------------------------------------------------------------------
// Fully specialized per-scale pipeline. All dims constexpr -> WMMA K-chains
// are straight-line, operand loads from LDS are unguarded.
// ---------------------------------------------------------------------------
template <int I>
__device__ __forceinline__ void process_scale(
    int t, int b, int m0, int lane, int waveId, int r16, int khalf, int X,
    const float* __restrict__ Amat, const float* __restrict__ ahat,
    const float* __restrict__ beta,
    float* __restrict__ sG, float* __restrict__ sUB, float* __restrict__ sA,
    const float* __restrict__ sC, const float* __restrict__ sS,
    float (&acc)[64]) {
    constexpr int c = SC_C[I];
    constexpr int cpad = (c + 3) & ~3;      // K padded to multiple of 4 (scale 5: 8)
    constexpr int h = SC_H[I];
    constexpr int w = h;
    constexpr int hw = h * w;
    constexpr int off = SC_OFF[I];
    constexpr int Lc = SC_LC[I];
    constexpr int twoLc = 2 * Lc;
    constexpr int ncols = h * twoLc;        // <= 256, power of two
    constexpr int nTiles = (ncols + 15) / 16;
    constexpr int nChunks = w / Lc;
    constexpr int rh = 32 / h;
    constexpr int lgr = (rh == 32) ? 5 : (rh == 16) ? 4 : (rh == 8) ? 3
                      : (rh == 4) ? 2 : (rh == 2) ? 1 : 0;
    constexpr int hLc = h * Lc;
    constexpr int AS = cpad + 1;            // padded sA row stride (odd)

    const float* ahatB = ahat + AHAT_BASE[I] + (size_t)b * (c * hw * 2);
    const float* betaB = beta + BETA_BASE[I];

    // stage A tile (16 x cpad), zero-padded K, padded stride vs bank conflicts
    for (int idx = t; idx < 16 * cpad; idx += 256) {
        int rr = idx / cpad;
        int cc = idx - rr * cpad;
        sA[rr * AS + cc] = (cc < c) ? Amat[(size_t)(m0 + rr) * 570 + off + cc] : 0.f;
    }
    __syncthreads();

#pragma unroll 1
    for (int chunk = 0; chunk < nChunks; chunk++) {
        const int l0 = chunk * Lc;

        // --- stage B panel (cpad x ncols) into sUB, zero-padded K ---
        for (int idx = t; idx < cpad * ncols; idx += 256) {
            int kk = idx / ncols;
            int colLocal = idx - kk * ncols;
            int kcol = colLocal / twoLc;
            int within = colLocal - kcol * twoLc;
            int n = (kcol * w + l0 + (within >> 1)) * 2 + (within & 1);
            sUB[idx] = (kk < c) ? ahatB[kk * (2 * hw) + n] : 0.f;
        }
        __syncthreads();

        // --- GEMM: G(16 x ncols) = Atile * Bpanel via WMMA f32 16x16x4 ---
#pragma unroll
        for (int tt = 0; tt < (nTiles + 7) / 8; tt++) {
            int tIdx = waveId + tt * 8;
            if (tIdx < nTiles) {                 // wave-uniform; folds when nTiles%8==0
                v8f cf = {0.f, 0.f, 0.f, 0.f, 0.f, 0.f, 0.f, 0.f};
                int colIdx = (tIdx << 4) + r16;
                int colC = (ncols >= 16) ? colIdx
                                         : ((colIdx < ncols) ? colIdx : (ncols - 1));
                const float* aRow = sA + r16 * AS + (khalf << 1);
                const float* bCol = sUB + colC + (khalf << 1) * ncols;
#pragma unroll
                for (int k0 = 0; k0 < cpad; k0 += 4) {
                    v2f af = {aRow[k0], aRow[k0 + 1]};
                    v2f bf = {bCol[k0 * ncols], bCol[(k0 + 1) * ncols]};
                    cf = __builtin_amdgcn_wmma_f32_16x16x4_f32(false, af, false, bf,
                                                               (short)0, cf, false, false);
                }
                if (ncols >= 16 || colIdx < ncols) {
#pragma unroll
                    for (int g = 0; g < 8; g++)
                        sG[(khalf * 8 + g) * ncols + colIdx] = cf[g];
                }
            }
        }
        __syncthreads();

        // --- beta multiply in place: H = G * beta ---
        constexpr int nElem = 16 * hLc;
        for (int idx = t; idx < nElem; idx += 256) {
            int r = idx / hLc;
            int rem = idx - r * hLc;
            int k = rem / Lc;
            int lc = rem - k * Lc;
            int ci = r * ncols + k * twoLc + (lc << 1);
            float gre = sG[ci], gim = sG[ci + 1];
            const float* bb = betaB + (size_t)((m0 + r) * hw + k * w + l0 + lc) * 2;
            float bre = bb[0], bim = bb[1];
            sG[ci] = gre * bre - gim * bim;
            sG[ci + 1] = gre * bim + gim * bre;
        }
        __syncthreads();

        // --- column IDFT: U[r,y,lc] = sum_k H[r,k,lc] e^{+2pi i k y / h} ---
        if constexpr (I >= 3) {
            // As a real WMMA GEMM: D[y,(r,lc)] = sum_kk E[y,kk]*B[kk,(r,lc)],
            // E rows interleave (cos,-sin) for U_re / (sin,cos) for U_im;
            // B rows interleave (H_re,H_im). K = 2h, N = 16*Lc, M = h.
            constexpr int kSteps = h / 2;            // K/4 = 2h/4
            constexpr int mTiles = (h + 15) / 16;
            constexpr int nTilesU = Lc;              // N/16 = 16*Lc/16
            constexpr int totTiles = 2 * mTiles * nTilesU;  // == 16 for I=3,4,5
#pragma unroll
            for (int tt = 0; tt < (totTiles + 7) / 8; tt++) {
                int tid = waveId + tt * 8;
                if (tid < totTiles) {                // wave-uniform; folds (totTiles==16)
                    int reimSel = tid & 1;
                    int mt = (tid >> 1) % mTiles;
                    int nt = (tid >> 1) / mTiles;
                    int y = mt * 16 + r16;           // A-row / D-row for this lane
                    int col = (nt << 4) + r16;       // B-col / D-col for this lane
                    int rU = col / Lc;
                    int lcU = col & (Lc - 1);
                    const float* bBase = sG + rU * ncols + (lcU << 1);
                    v8f cf = {0.f, 0.f, 0.f, 0.f, 0.f, 0.f, 0.f, 0.f};
#pragma unroll
                    for (int ks = 0; ks < kSteps; ks++) {
                        int k = (ks << 1) + khalf;
                        int tw = (k * y * rh) & 31;
                        float cv = sC[tw], sv = sS[tw];
                        v2f af;
                        if (reimSel == 0) { af.x = cv; af.y = -sv; }
                        else              { af.x = sv; af.y = cv; }
                        v2f bf = {bBase[k * twoLc], bBase[k * twoLc + 1]};
                        cf = __builtin_amdgcn_wmma_f32_16x16x4_f32(false, af, false, bf,
                                                                   (short)0, cf, false, false);
                    }
#pragma unroll
                    for (int g = 0; g < 8; g++) {
                        int yy = mt * 16 + khalf * 8 + g;
                        if (h >= 16 || yy < h)        // folds for I=4,5
                            sUB[(((rU * h + yy) * Lc + lcU) << 1) + reimSel] = cf[g];
                    }
                }
            }
        } else {
            for (int idx = t; idx < nElem; idx += 256) {
                int r = idx / hLc;
                int rem = idx - r * hLc;
                int y = rem / Lc;
                int lc = rem - y * Lc;
                float ure = 0.f, uim = 0.f;
                int base = r * ncols + (lc << 1);
#pragma unroll
                for (int k = 0; k < h; k++) {
                    float hre = sG[base + k * twoLc];
                    float him = sG[base + k * twoLc + 1];
                    int tw = (k * y * rh) & 31;
                    float ct = sC[tw], st = sS[tw];
                    ure += hre * ct - him * st;
                    uim += hre * st + him * ct;
                }
                int ui = ((r * h + y) * Lc + lc) << 1;
                sUB[ui] = ure;
                sUB[ui + 1] = uim;
            }
        }
        __syncthreads();

        // --- row IDFT (partial over this l-chunk) into fixed-ownership acc ---
        if ((X & (rh - 1)) == 0) {
            int x = X >> lgr;
#pragma unroll
            for (int j = 0; j < 64; j++) {
                int e = t + (j << 8);
                int Y = (e >> 5) & 31;
                if ((Y & (rh - 1)) == 0) {
                    int y = Y >> lgr;
                    int r = e >> 10;
                    float s = 0.f;
                    int ub = ((r * h + y) * Lc) << 1;
#pragma unroll
                    for (int lc = 0; lc < Lc; lc++) {
                        int tw = ((l0 + lc) * x * rh) & 31;
                        s += sUB[ub + (lc << 1)] * sC[tw] - sUB[ub + (lc << 1) + 1] * sS[tw];
                    }
                    acc[j] += s;
                }
            }
        }
        __syncthreads();
    }
}

// ---------------------------------------------------------------------------
// Kernel 3: fused per-(b, 16-m tile) pipeline over all 6 scales.
// ---------------------------------------------------------------------------
__launch_bounds__(256)
__global__ void mega_kernel(const float* __restrict__ Amat, const float* __restrict__ ahat,
                            const float* __restrict__ beta, float* __restrict__ out) {
    __shared__ float sG[16 * 256];     // GEMM output / H tile (16 rows x <=256 cols)
    __shared__ float sUB[4096];        // B panel during GEMM, then column-IDFT result
    __shared__ float sA[16 * 385];     // padded A tile (max cpad=384, stride cpad+1)
    __shared__ float sC[32], sS[32];   // e^{+2pi i t/32}

    const int t = threadIdx.x;
    const int b = blockIdx.x >> 6;
    const int m0 = (blockIdx.x & 63) << 4;
    const int lane = t & 31;
    const int waveId = t >> 5;
    const int r16 = lane & 15;
    const int khalf = lane >> 4;
    const int X = t & 31;

    if (t < 32) {
        float ang = (float)t * TWO_PI_OVER_32;
        sC[t] = cosf(ang);
        sS[t] = sinf(ang);
    }

    float acc[64];
#pragma unroll
    for (int j = 0; j < 64; j++) acc[j] = 0.f;
    __syncthreads();

    process_scale<0>(t, b, m0, lane, waveId, r16, khalf, X, Amat, ahat, beta, sG, sUB, sA, sC, sS, acc);
    process_scale<1>(t, b, m0, lane, waveId, r16, khalf, X, Amat, ahat, beta, sG, sUB, sA, sC, sS, acc);
    process_scale<2>(t, b, m0, lane, waveId, r16, khalf, X, Amat, ahat, beta, sG, sUB, sA, sC, sS, acc);
    process_scale<3>(t, b, m0, lane, waveId, r16, khalf, X, Amat, ahat, beta, sG, sUB, sA, sC, sS, acc);
    process_scale<4>(t, b, m0, lane, waveId, r16, khalf, X, Amat, ahat, beta, sG, sUB, sA, sC, sS, acc);
    process_scale<5>(t, b, m0, lane, waveId, r16, khalf, X, Amat, ahat, beta, sG, sUB, sA, sC, sS, acc);

    // single coalesced store of the exclusive out tile; 1/1024 = ifft2 norm
    size_t outBase = ((size_t)b << 20) + ((size_t)m0 << 10);
#pragma unroll
    for (int j = 0; j < 64; j++)
        out[outBase + t + (j << 8)] = acc[j] * (1.0f / 1024.0f);
}

// ---------------------------------------------------------------------------
extern "C" void kernel_launch(void* const* d_in, const int* in_sizes, int n_in,
                              void* d_out, int out_size, void* d_ws, size_t ws_size,
                              hipStream_t stream) {
    (void)in_sizes; (void)n_in; (void)out_size; (void)ws_size;
    const float* a0 = (const float*)d_in[0];
    const float* a1 = (const float*)d_in[1];
    const float* a2 = (const float*)d_in[2];
    const float* a3 = (const float*)d_in[3];
    const float* a4 = (const float*)d_in[4];
    const float* a5 = (const float*)d_in[5];
    const float* A = (const float*)d_in[6];
    const float* alpha = (const float*)d_in[7];
    float* out = (float*)d_out;

    float* ws = (float*)d_ws;
    float* ahat = ws;                       // 786432 floats (3 MB)
    float* beta = ws + AHAT_TOTAL_FLOATS;   // 2795520 floats (10.7 MB)

    dft2_kernel<<<AHAT_TOTAL_CPLX / 256, 256, 0, stream>>>(a0, a1, a2, a3, a4, a5, ahat);
    beta_kernel<<<BETA_TOTAL_CPLX / 256, 256, 0, stream>>>(alpha, beta);
    mega_kernel<<<2048, 256, 0, stream>>>(A, ahat, beta, out);
}